// ControlledSystem_73504070304276
// MI455X (gfx1250) — compile-verified
//
#include <hip/hip_runtime.h>

// CDNA5 / gfx1250: wave32, WMMA f32 16x16x4 for the MLP second layer (full f32
// precision), hardware tanh/exp2/log2 for activations. Two independent WMMA
// accumulation chains (one per 16-row tile) are interleaved in a single K-loop
// so tanh (TRANS) work co-executes under the matrix pipe and the W1/b1 loads
// are shared between tiles.

typedef float v2f __attribute__((ext_vector_type(2)));
typedef float v8f __attribute__((ext_vector_type(8)));

#define BLOCK 256  // 8 waves per workgroup

static __device__ __forceinline__ float fast_tanh(float x) {
#if __has_builtin(__builtin_amdgcn_tanhf)
  return __builtin_amdgcn_tanhf(x);          // v_tanh_f32 (gfx1250 TRANS op)
#elif __has_builtin(__builtin_amdgcn_tanh_f32)
  return __builtin_amdgcn_tanh_f32(x);
#else
  // tanh(x) = 1 - 2/(e^{2x}+1), e^{2x} = 2^{2x*log2(e)}  (v_exp_f32)
  float e = __builtin_amdgcn_exp2f(x * 2.8853900817779268f);
  return 1.0f - 2.0f / (e + 1.0f);
#endif
}

static __device__ __forceinline__ float softplus_f(float x) {
  // stable: max(x,0) + ln(1 + exp(-|x|)); v_exp_f32/v_log_f32 are base-2
  float t = __builtin_amdgcn_exp2f(-__builtin_fabsf(x) * 1.4426950408889634f);
  return fmaxf(x, 0.0f) + 0.6931471805599453f * __builtin_amdgcn_logf(1.0f + t);
}

__global__ __launch_bounds__(BLOCK) void csys_kernel(
    const float* __restrict__ t_in, const float* __restrict__ z,
    const float* __restrict__ logK, const float* __restrict__ logz,
    const float* __restrict__ logp, const float* __restrict__ W1,
    const float* __restrict__ b1, const float* __restrict__ W2,
    const float* __restrict__ b2, float* __restrict__ out, int n) {
  // per-wave exchange buffer: 32 elements x 2 MLP outputs
  __shared__ float xch[2 * BLOCK];

  const int tid   = threadIdx.x;
  const int lane  = tid & 31;
  const int wbase = (tid >> 5) * 64;
  const int gid   = blockIdx.x * BLOCK + tid;
  // clamp instead of early-exit: EXEC must stay all-ones for WMMA.
  // (clamped tail lanes recompute element n-1 and store identical values)
  const int idx = (gid < n) ? gid : (n - 1);

  // ---- coalesced state loads -------------------------------------------
  const float tt = t_in[idx];
  const float x1 = z[0 * n + idx];
  const float v1 = z[1 * n + idx];
  const float x2 = z[2 * n + idx];
  const float v2 = z[3 * n + idx];
  const float xc = z[4 * n + idx];

  const int khalf = lane >> 4;  // 0: K pair {0,1} of chunk, 1: pair {2,3}
  const int nn    = lane & 15;  // A: row M / B: col N / C/D: col N

  // ---- B matrix tiles for W2 [64][2], padded N=2 -> 16 (shared by tiles) --
  v2f bmat[16];
#pragma unroll
  for (int c = 0; c < 16; ++c) {
    const int kb = 4 * c + 2 * khalf;
    float bx = 0.0f, by = 0.0f;
    if (nn < 2) {
      bx = W2[kb * 2 + nn];
      by = W2[(kb + 1) * 2 + nn];
    }
    bmat[c].x = bx;
    bmat[c].y = by;
  }

  // ---- MLP: h = tanh(v2*W1+b1) produced directly in WMMA-A layout; both
  // ---- 16-row tiles advance together: two independent chains of 16
  // ---- V_WMMA_F32_16X16X4_F32 sharing W1/b1 loads and the B operand.
  const float v2t0 = __shfl(v2, nn, 32);       // tile 0: rows = lanes 0..15
  const float v2t1 = __shfl(v2, 16 + nn, 32);  // tile 1: rows = lanes 16..31
  v8f acc0 = {};
  v8f acc1 = {};
#pragma unroll
  for (int c = 0; c < 16; ++c) {
    const int ka = 4 * c + 2 * khalf;
    const float2 w1p = *reinterpret_cast<const float2*>(W1 + ka);
    const float2 b1p = *reinterpret_cast<const float2*>(b1 + ka);
    v2f a0, a1;
    a0.x = fast_tanh(fmaf(v2t0, w1p.x, b1p.x));
    a0.y = fast_tanh(fmaf(v2t0, w1p.y, b1p.y));
    a1.x = fast_tanh(fmaf(v2t1, w1p.x, b1p.x));
    a1.y = fast_tanh(fmaf(v2t1, w1p.y, b1p.y));
    // D = A(16x4 f32) x B(4x16 f32) + C, f32 accumulate
    acc0 = __builtin_amdgcn_wmma_f32_16x16x4_f32(
        false, a0, false, bmat[c], (short)0, acc0, false, false);
    acc1 = __builtin_amdgcn_wmma_f32_16x16x4_f32(
        false, a1, false, bmat[c], (short)0, acc1, false, false);
  }

  // D layout: VGPR r, lanes 0-15 -> M=r, lanes 16-31 -> M=8+r, N=lane&15
  if (nn < 2) {
    const int wv0 = khalf * 8;       // tile 0 rows
    const int wv1 = 16 + khalf * 8;  // tile 1 rows
#pragma unroll
    for (int r = 0; r < 8; ++r) {
      xch[wbase + (wv0 + r) * 2 + nn] = acc0[r];
      xch[wbase + (wv1 + r) * 2 + nn] = acc1[r];
    }
  }
  __syncthreads();

  const float kinetic = softplus_f(xch[wbase + lane * 2 + 0] + b2[0]);
  const float stic    = softplus_f(xch[wbase + lane * 2 + 1] + b2[1]);

  // ---- elementwise physics ---------------------------------------------
  const float Kc = expf(logK[0]);
  const float zc = expf(logz[0]);
  const float pc = expf(logp[0]);

  const float x2ref = 0.5f * __sinf(0.5f * tt);
  const float e     = x2ref - x2;
  const float d_xc  = -pc * xc + e;
  const float u     = Kc * (zc - pc) * xc + Kc * e;

  const float dx1 = v1;
  // M1 = 1.0
  const float dv1 = u - 2.0f * x1 - 0.5f * v1 - 3.0f * (x1 - x2) - 0.8f * (v1 - v2);
  const float dx2 = v2;

  const float F_net    = 3.0f * (x1 - x2) + 0.8f * (v1 - v2);
  const float F_static = -fminf(fmaxf(F_net, -stic), stic);
  const float sgn      = (v2 > 0.0f) ? 1.0f : ((v2 < 0.0f) ? -1.0f : 0.0f);
  const float F_kin    = -kinetic * sgn;
  const float F_fric   = (__builtin_fabsf(v2) < 0.01f) ? F_static : F_kin;
  const float dv2      = (F_net + F_fric) * (1.0f / 1.5f);  // / M2

  out[0 * n + idx] = dx1;
  out[1 * n + idx] = dv1;
  out[2 * n + idx] = dx2;
  out[3 * n + idx] = dv2;
  out[4 * n + idx] = d_xc;
}

extern "C" void kernel_launch(void* const* d_in, const int* in_sizes, int n_in,
                              void* d_out, int out_size, void* d_ws,
                              size_t ws_size, hipStream_t stream) {
  const float* t    = (const float*)d_in[0];
  const float* z    = (const float*)d_in[1];
  const float* logK = (const float*)d_in[2];
  const float* logz = (const float*)d_in[3];
  const float* logp = (const float*)d_in[4];
  const float* W1   = (const float*)d_in[5];
  const float* b1   = (const float*)d_in[6];
  const float* W2   = (const float*)d_in[7];
  const float* b2   = (const float*)d_in[8];
  float* out        = (float*)d_out;

  const int n      = in_sizes[0];
  const int blocks = (n + BLOCK - 1) / BLOCK;
  hipLaunchKernelGGL(csys_kernel, dim3(blocks), dim3(BLOCK), 0, stream,
                     t, z, logK, logz, logp, W1, b1, W2, b2, out, n);
}